// PGCN_81449759801399
// MI455X (gfx1250) — compile-verified
//
#include <hip/hip_runtime.h>
#include <math.h>

typedef float v2f __attribute__((ext_vector_type(2)));
typedef float v8f __attribute__((ext_vector_type(8)));

#define ALPHA_MIX 0.1f

enum { MODE_STORE = 0, MODE_RELU_MIX = 1 };

__device__ __forceinline__ v8f wmma_f32(v2f a, v2f b, v8f c) {
    // D = A(16x4,f32) * B(4x16,f32) + C(16x16,f32)
    return __builtin_amdgcn_wmma_f32_16x16x4_f32(
        /*neg_a=*/false, a, /*neg_b=*/false, b,
        /*c_mod=*/(short)0, c, /*reuse_a=*/false, /*reuse_b=*/false);
}

// ---------------------------------------------------------------------------
// C(M x 128) = A(M x K) @ B(K x 128), fp32 WMMA 16x16x4.
// mode==MODE_RELU_MIX: C = 0.1*resid + 0.9*relu(acc)
// Block = 128 threads = 4 waves; each wave owns one 16-row strip and all 8
// 16-column tiles (whole 128-wide output row panel).
// ---------------------------------------------------------------------------
__global__ __launch_bounds__(128) void gemm128_wmma(
    const float* __restrict__ A, const float* __restrict__ B,
    float* __restrict__ C, const float* __restrict__ resid,
    int M, int K, int mode) {
    const int lane = threadIdx.x & 31;
    const int wave = threadIdx.x >> 5;
    const int m0 = (blockIdx.x * 4 + wave) * 16;
    if (m0 >= M) return;                       // whole-wave early exit: EXEC stays all-1s for WMMA
    const int h  = lane >> 4;                  // K-half selector (lanes 0-15 -> K 0,1; 16-31 -> K 2,3)
    const int mr = lane & 15;                  // row (A) / col (B) within tile

    v8f acc[8];
    for (int t = 0; t < 8; ++t)
        for (int i = 0; i < 8; ++i) acc[t][i] = 0.0f;

    const float* Arow = A + (size_t)(m0 + mr) * K;
    for (int k0 = 0; k0 < K; k0 += 4) {
        v2f a;
        a.x = Arow[k0 + 2 * h];
        a.y = Arow[k0 + 2 * h + 1];
        const float* B0 = B + (size_t)(k0 + 2 * h) * 128;
        const float* B1 = B0 + 128;
#pragma unroll
        for (int t = 0; t < 8; ++t) {
            v2f b;
            b.x = B0[16 * t + mr];
            b.y = B1[16 * t + mr];
            acc[t] = wmma_f32(a, b, acc[t]);
        }
    }

#pragma unroll
    for (int t = 0; t < 8; ++t) {
#pragma unroll
        for (int r = 0; r < 8; ++r) {
            const int row = m0 + r + 8 * h;    // C layout: VGPR r -> M=r (lanes<16), M=r+8 (lanes>=16)
            const int col = 16 * t + mr;
            float v = acc[t][r];
            if (mode == MODE_RELU_MIX)
                v = ALPHA_MIX * resid[(size_t)row * 128 + col]
                  + (1.0f - ALPHA_MIX) * fmaxf(v, 0.0f);
            C[(size_t)row * 128 + col] = v;
        }
    }
}

// ---------------------------------------------------------------------------
// r[n,d] = (1/P) * sum_{p,l} feats[paths[p,n,l], d] * pw[l,d]
// block = node n, thread = feature d (D=128). Memory bound; feats reads coalesce.
// ---------------------------------------------------------------------------
__global__ __launch_bounds__(128) void path_agg(
    const float* __restrict__ feats, const int* __restrict__ paths,
    const float* __restrict__ pw, float* __restrict__ r,
    int N, int P, int L) {
    const int n = blockIdx.x;
    const int d = threadIdx.x;
    float acc = 0.0f;
    for (int p = 0; p < P; ++p) {
        const int* prow = paths + ((size_t)p * N + n) * L;
        for (int l = 0; l < L; ++l) {
            const int idx = prow[l];
            acc += feats[(size_t)idx * 128 + d] * pw[l * 128 + d];
        }
    }
    r[(size_t)n * 128 + d] = acc * (1.0f / (float)P);
}

// ---------------------------------------------------------------------------
// Fused per-16-sample-tile: embedding gather -> (512->256) -> (256->64) -> (64->1)+sigmoid.
// No activations between MLP linears (matches reference). LDS strides padded
// (516/260/68 == 4 mod 64 banks) so WMMA A-fragment reads are conflict-free.
// 4 waves: wave w owns x0 n-tiles {w, w+4, w+8, w+12} and x1 n-tile w.
// ---------------------------------------------------------------------------
__global__ __launch_bounds__(128) void mlp_fused(
    const float* __restrict__ feats_mm, const float* __restrict__ feats_dd,
    const float* __restrict__ feats_md, const int* __restrict__ samples,
    const float* __restrict__ W0, const float* __restrict__ b0,
    const float* __restrict__ W1, const float* __restrict__ b1,
    const float* __restrict__ W2, const float* __restrict__ b2,
    float* __restrict__ out, int Nm) {
    __shared__ float Ae[16 * 516];
    __shared__ float X0[16 * 260];
    __shared__ float X1[16 * 68];

    const int tid  = threadIdx.x;
    const int lane = tid & 31;
    const int wave = tid >> 5;
    const int s0   = blockIdx.x * 16;

    // Stage 1: gather emb tile (16 x 512): [mm[i0] | md[i0] | dd[i1] | md[Nm+i1]]
    for (int idx = tid; idx < 16 * 512; idx += 128) {
        const int s = idx >> 9, c = idx & 511;
        const int i0 = samples[2 * (s0 + s) + 0];
        const int i1 = samples[2 * (s0 + s) + 1];
        float v;
        if (c < 128)      v = feats_mm[(size_t)i0 * 128 + c];
        else if (c < 256) v = feats_md[(size_t)i0 * 128 + (c - 128)];
        else if (c < 384) v = feats_dd[(size_t)i1 * 128 + (c - 256)];
        else              v = feats_md[(size_t)(Nm + i1) * 128 + (c - 384)];
        Ae[s * 516 + c] = v;
    }
    __syncthreads();

    const int h  = lane >> 4;
    const int mr = lane & 15;

    // Stage 2: X0(16x256) = Ae(16x512) @ W0(512x256) + b0
    {
        v8f acc[4];
        for (int u = 0; u < 4; ++u)
            for (int i = 0; i < 8; ++i) acc[u][i] = 0.0f;
        for (int k0 = 0; k0 < 512; k0 += 4) {
            v2f a;
            a.x = Ae[mr * 516 + k0 + 2 * h];
            a.y = Ae[mr * 516 + k0 + 2 * h + 1];
#pragma unroll
            for (int u = 0; u < 4; ++u) {
                const int n = (wave + 4 * u) * 16 + mr;
                v2f b;
                b.x = W0[(k0 + 2 * h) * 256 + n];
                b.y = W0[(k0 + 2 * h + 1) * 256 + n];
                acc[u] = wmma_f32(a, b, acc[u]);
            }
        }
#pragma unroll
        for (int u = 0; u < 4; ++u)
#pragma unroll
            for (int r = 0; r < 8; ++r) {
                const int row = r + 8 * h;
                const int col = (wave + 4 * u) * 16 + mr;
                X0[row * 260 + col] = acc[u][r] + b0[col];
            }
    }
    __syncthreads();

    // Stage 3: X1(16x64) = X0(16x256) @ W1(256x64) + b1
    {
        v8f acc;
        for (int i = 0; i < 8; ++i) acc[i] = 0.0f;
        for (int k0 = 0; k0 < 256; k0 += 4) {
            v2f a;
            a.x = X0[mr * 260 + k0 + 2 * h];
            a.y = X0[mr * 260 + k0 + 2 * h + 1];
            const int n = wave * 16 + mr;
            v2f b;
            b.x = W1[(k0 + 2 * h) * 64 + n];
            b.y = W1[(k0 + 2 * h + 1) * 64 + n];
            acc = wmma_f32(a, b, acc);
        }
#pragma unroll
        for (int r = 0; r < 8; ++r) {
            const int row = r + 8 * h;
            const int col = wave * 16 + mr;
            X1[row * 68 + col] = acc[r] + b1[col];
        }
    }
    __syncthreads();

    // Stage 4: out = sigmoid(X1 @ W2 + b2)
    if (tid < 16) {
        float dot = b2[0];
        for (int j = 0; j < 64; ++j) dot += X1[tid * 68 + j] * W2[j];
        out[s0 + tid] = 1.0f / (1.0f + expf(-dot));
    }
}

// ---------------------------------------------------------------------------
extern "C" void kernel_launch(void* const* d_in, const int* in_sizes, int n_in,
                              void* d_out, int out_size, void* d_ws, size_t ws_size,
                              hipStream_t stream) {
    const int*   paths_mm = (const int*)d_in[0];
    const int*   paths_dd = (const int*)d_in[1];
    const int*   paths_md = (const int*)d_in[2];
    const float* miRNA    = (const float*)d_in[3];
    const float* disease  = (const float*)d_in[4];
    const int*   samples  = (const int*)d_in[5];
    const float* Wm  = (const float*)d_in[6];
    const float* Wd  = (const float*)d_in[7];
    const float* pw1 = (const float*)d_in[8];
    const float* pw2 = (const float*)d_in[9];
    const float* fcW = (const float*)d_in[10];
    const float* W0  = (const float*)d_in[11];
    const float* b0  = (const float*)d_in[12];
    const float* W1  = (const float*)d_in[13];
    const float* b1  = (const float*)d_in[14];
    const float* W2  = (const float*)d_in[15];
    const float* b2  = (const float*)d_in[16];

    const int Nm = 8000, Nd = 6000, D = 128, P = 8, L1 = 4, L2 = 6;
    const int Nmd = Nm + Nd;
    const int S = 200000;

    // Workspace layout (floats): ~36 MB total
    float* ws       = (float*)d_ws;
    float* init_m   = ws;                            // Nm*D
    float* init_d   = init_m  + (size_t)Nm  * D;     // Nd*D
    float* init_md  = init_d  + (size_t)Nd  * D;     // Nmd*D
    float* feats_mm = init_md + (size_t)Nmd * D;     // Nm*D
    float* feats_dd = feats_mm + (size_t)Nm * D;     // Nd*D
    float* feats_md = feats_dd + (size_t)Nd * D;     // Nmd*D
    float* rbuf     = feats_md + (size_t)Nmd * D;    // Nmd*D

    auto gemm = [&](const float* A, const float* B, float* C, const float* resid,
                    int M, int K, int mode) {
        const int blocks = (M / 16 + 3) / 4;
        hipLaunchKernelGGL(gemm128_wmma, dim3(blocks), dim3(128), 0, stream,
                           A, B, C, resid, M, K, mode);
    };

    // init_m = miRNA @ Wm ; init_d = disease @ Wd
    gemm(miRNA,   Wm, init_m, nullptr, Nm, Nm, MODE_STORE);
    gemm(disease, Wd, init_d, nullptr, Nd, Nd, MODE_STORE);

    // init_md = concat(init_m, init_d)
    hipMemcpyAsync(init_md, init_m, (size_t)Nm * D * sizeof(float),
                   hipMemcpyDeviceToDevice, stream);
    hipMemcpyAsync(init_md + (size_t)Nm * D, init_d, (size_t)Nd * D * sizeof(float),
                   hipMemcpyDeviceToDevice, stream);

    // mm path layers
    for (int l = 0; l < 2; ++l) {
        const float* src = (l == 0) ? init_m : feats_mm;
        path_agg<<<Nm, 128, 0, stream>>>(src, paths_mm, pw1 + (size_t)l * L1 * D,
                                         rbuf, Nm, P, L1);
        gemm(rbuf, fcW + (size_t)l * D * D, feats_mm, init_m, Nm, D, MODE_RELU_MIX);
    }
    // dd path layers
    for (int l = 0; l < 2; ++l) {
        const float* src = (l == 0) ? init_d : feats_dd;
        path_agg<<<Nd, 128, 0, stream>>>(src, paths_dd, pw1 + (size_t)l * L1 * D,
                                         rbuf, Nd, P, L1);
        gemm(rbuf, fcW + (size_t)l * D * D, feats_dd, init_d, Nd, D, MODE_RELU_MIX);
    }
    // md path layers
    for (int l = 0; l < 2; ++l) {
        const float* src = (l == 0) ? init_md : feats_md;
        path_agg<<<Nmd, 128, 0, stream>>>(src, paths_md, pw2 + (size_t)l * L2 * D,
                                          rbuf, Nmd, P, L2);
        gemm(rbuf, fcW + (size_t)l * D * D, feats_md, init_md, Nmd, D, MODE_RELU_MIX);
    }

    // fused embedding gather + MLP + sigmoid
    mlp_fused<<<S / 16, 128, 0, stream>>>(feats_mm, feats_dd, feats_md, samples,
                                          W0, b0, W1, b1, W2, b2,
                                          (float*)d_out, Nm);
}